// BoundaryLoss_41798621725152
// MI455X (gfx1250) — compile-verified
//
#include <hip/hip_runtime.h>

#define THREADS 256

typedef __attribute__((ext_vector_type(2))) float v2f;
typedef __attribute__((ext_vector_type(4))) float v4f;
typedef __attribute__((ext_vector_type(8))) float v8f;

namespace {
constexpr int RG    = 96;          // tile + 2*16 halo
constexpr int TILE  = 64;
constexpr int HALO  = 16;
constexpr int IMG   = 512;
constexpr int NCLS  = 8;
constexpr int HWSZ  = IMG * IMG;
constexpr int ITERS = 15;
constexpr int W3    = 3;           // 96 bits = 3 dwords per row
constexpr int NW    = RG * W3;     // 288 words per bitmap
}

__device__ __forceinline__ void async_cp_b32(const int* g, int* l) {
#if __has_builtin(__builtin_amdgcn_global_load_async_to_lds_b32)
  int* gnc = const_cast<int*>(g);
  __builtin_amdgcn_global_load_async_to_lds_b32(
      (__attribute__((address_space(1))) int*)gnc,
      (__attribute__((address_space(3))) int*)l, 0, 0);
#else
  *l = *g;
#endif
}

__device__ __forceinline__ void wait_async_lds() {
#if __has_builtin(__builtin_amdgcn_global_load_async_to_lds_b32)
 #if __has_builtin(__builtin_amdgcn_s_wait_asynccnt)
  __builtin_amdgcn_s_wait_asynccnt(0);
 #else
  asm volatile("s_wait_asynccnt 0" ::: "memory");
 #endif
#endif
}

// Deterministic 256-thread block reduction. All 8 waves run the WMMA with
// EXEC all-ones (redundantly); A holds the 32 lane-partials (K=0 / K=2
// columns), B = ones => row m of D = s[m] + s[m+16]; per-lane sum of the 8
// D VGPRs + xor-16 shuffle yields the full 32-lane total on every lane.
__device__ __forceinline__ void block_reduce_wmma(float v, float scale,
                                                  float* dst, float* s_red) {
  const int tid = threadIdx.x;
  s_red[tid] = v;
  __syncthreads();
  const int lane = tid & 31;
  float s = 0.f;
#pragma unroll
  for (int k = 0; k < 8; ++k) s += s_red[lane + 32 * k];
  v2f a; a[0] = s;   a[1] = 0.f;
  v2f b; b[0] = 1.f; b[1] = 1.f;
  v8f c = {};
  v8f d = __builtin_amdgcn_wmma_f32_16x16x4_f32(false, a, false, b,
                                                (short)0, c, false, false);
  float t = d[0] + d[1] + d[2] + d[3] + d[4] + d[5] + d[6] + d[7];
  t += __shfl_xor(t, 16, 32);
  if (tid == 0) *dst = t * scale;
}

__global__ __launch_bounds__(THREADS)
void boundary_loss_tile(const float* __restrict__ inputs,
                        const int* __restrict__ targets,
                        float* __restrict__ partial) {
  __shared__ int           s_t [RG * RG];   // clamped target window
  __shared__ unsigned char s_hx[RG * RG];   // packed h-min|h-max nibbles
  __shared__ unsigned char s_d [RG * RG];   // capped Chebyshev distance
  __shared__ unsigned      s_m0[NW];        // mask bitmap (cur)
  __shared__ unsigned      s_m1[NW];        // mask bitmap (h-dilated)
  __shared__ float         s_red[THREADS];

  const int tid  = threadIdx.x;
  const int blk  = blockIdx.x;
  const int bimg = blk >> 6;                // 16 images x 64 tiles
  const int tix  = blk & 63;
  const int oy   = (tix >> 3) * TILE;
  const int ox   = (tix & 7) * TILE;

  // ---- stage 1: async-stage clamped 96x96 target window into LDS ----
  const int* tbase = targets + bimg * HWSZ;
  for (int idx = tid; idx < RG * RG; idx += THREADS) {
    int ly = idx / RG, lx = idx - ly * RG;
    int gy = oy - HALO + ly; gy = gy < 0 ? 0 : (gy > IMG - 1 ? IMG - 1 : gy);
    int gx = ox - HALO + lx; gx = gx < 0 ? 0 : (gx > IMG - 1 ? IMG - 1 : gx);
    async_cp_b32(tbase + gy * IMG + gx, &s_t[idx]);
  }
  wait_async_lds();
  __syncthreads();

  // ---- stage 2: horizontal class min/max, packed in nibbles ----
  for (int idx = tid; idx < RG * RG; idx += THREADS) {
    int lx = idx % RG;
    int xm = (lx > 0)      ? idx - 1 : idx;
    int xp = (lx < RG - 1) ? idx + 1 : idx;
    int t0 = s_t[xm], t1 = s_t[idx], t2 = s_t[xp];
    int mn = min(t0, min(t1, t2));
    int mx = max(t0, max(t1, t2));
    s_hx[idx] = (unsigned char)(mn | (mx << 4));
    s_d[idx]  = 0;
  }
  __syncthreads();

  // ---- stage 3: vertical pass -> boundary bitmap (min3x3 != max3x3) ----
  for (int w = tid; w < NW; w += THREADS) {
    int ly = w / W3, wi = w - ly * W3;
    unsigned bits = 0;
    if (ly >= 1 && ly <= RG - 2) {
#pragma unroll
      for (int b2 = 0; b2 < 32; ++b2) {
        int lx = wi * 32 + b2;
        if (lx >= 1 && lx <= RG - 2) {
          int idx = ly * RG + lx;
          int a0 = s_hx[idx - RG], a1 = s_hx[idx], a2 = s_hx[idx + RG];
          int mn = min(a0 & 15, min(a1 & 15, a2 & 15));
          int mx = max(a0 >> 4, max(a1 >> 4, a2 >> 4));
          bits |= (unsigned)(mn != mx) << b2;
        }
      }
    }
    s_m0[w] = bits;
  }
  __syncthreads();

  // ---- stage 4: 15 binary dilations (3x3 maxpool of a 0/1 mask == OR on
  //      bitmaps); record first-on iteration = capped Chebyshev dist ----
  for (int it = 0; it < ITERS; ++it) {
    for (int w = tid; w < NW; w += THREADS) {
      int wi = w % W3;
      unsigned h   = s_m0[w];
      unsigned lft = (wi > 0)      ? s_m0[w - 1] : 0u;
      unsigned rgt = (wi < W3 - 1) ? s_m0[w + 1] : 0u;
      s_m1[w] = h | (h << 1) | (h >> 1) | (lft >> 31) | (rgt << 31);
    }
    __syncthreads();
    for (int w = tid; w < NW; w += THREADS) {
      int ly = w / W3, wi = w - ly * W3;
      unsigned v = s_m1[w];
      if (ly > 0)      v |= s_m1[w - W3];
      if (ly < RG - 1) v |= s_m1[w + W3];
      unsigned old  = s_m0[w];
      unsigned diff = v & ~old;
      while (diff) {
        int b2 = __ffs(diff) - 1;
        s_d[ly * RG + wi * 32 + b2] = (unsigned char)(it + 1);
        diff &= diff - 1;
      }
      s_m0[w] = v;
    }
    __syncthreads();
  }

  // ---- stage 5: weighted cross-entropy, 4 pixels/thread-iter via b128 ----
  const float* ibase = inputs + (size_t)bimg * NCLS * HWSZ;
  float lsum = 0.f;
  for (int q = tid; q < (TILE * TILE / 4); q += THREADS) {
    int p  = q * 4;                      // 4 consecutive pixels, same row,
    int hy = p >> 6, hx = p & 63;        // 16B-aligned (ox, hx % 4 == 0)
    size_t off = (size_t)(oy + hy) * IMG + (ox + hx);
    v4f xv[NCLS];
#pragma unroll
    for (int c = 0; c < NCLS; ++c)
      xv[c] = *(const v4f*)(ibase + (size_t)c * HWSZ + off);
#pragma unroll
    for (int j = 0; j < 4; ++j) {
      int lidx = (hy + HALO) * RG + (hx + j + HALO);
      int tgt  = s_t[lidx];
      float m = xv[0][j];
#pragma unroll
      for (int c = 1; c < NCLS; ++c) m = fmaxf(m, xv[c][j]);
      float se = 0.f;
#pragma unroll
      for (int c = 0; c < NCLS; ++c) se += __expf(xv[c][j] - m);
      float lse = m + __logf(se);
      float xt = xv[0][j];
#pragma unroll
      for (int c = 1; c < NCLS; ++c) xt = (tgt == c) ? xv[c][j] : xt;
      float wgt = __expf((float)s_d[lidx] * (-1.f / 5.f));
      lsum += wgt * (lse - xt);
    }
  }

  block_reduce_wmma(lsum, 1.0f, &partial[blk], s_red);
}

__global__ __launch_bounds__(THREADS)
void boundary_loss_final(const float* __restrict__ partial,
                         float* __restrict__ out) {
  __shared__ float s_red[THREADS];
  const int tid = threadIdx.x;
  float s = 0.f;
#pragma unroll
  for (int k = 0; k < 4; ++k) s += partial[tid + THREADS * k];
  block_reduce_wmma(s, 1.0f / 4194304.0f, out, s_red);  // 1/(16*512*512), exact pow2
}

extern "C" void kernel_launch(void* const* d_in, const int* in_sizes, int n_in,
                              void* d_out, int out_size, void* d_ws, size_t ws_size,
                              hipStream_t stream) {
  const float* inputs  = (const float*)d_in[0];   // [16,8,512,512] f32
  const int*   targets = (const int*)d_in[1];     // [16,512,512] i32
  float* out     = (float*)d_out;                 // scalar
  float* partial = (float*)d_ws;                  // 1024 block partials

  boundary_loss_tile<<<dim3(1024), dim3(THREADS), 0, stream>>>(inputs, targets, partial);
  boundary_loss_final<<<dim3(1), dim3(THREADS), 0, stream>>>(partial, out);
}